// GraphAttentionEmbedding_70669391888433
// MI455X (gfx1250) — compile-verified
//
#include <hip/hip_runtime.h>
#include <math.h>

#define NN 50000
#define NE 500000

typedef float v2f __attribute__((ext_vector_type(2)));
typedef float v8f __attribute__((ext_vector_type(8)));

// ---------------- fill ----------------
__global__ void fill_kernel(float* __restrict__ p, float val, long n) {
  long i = (long)blockIdx.x * blockDim.x + threadIdx.x;
  if (i < n) p[i] = val;
}

// ---------------- fp32 WMMA GEMM: C[M,128] = A[M,K] @ B[K,128] + bias ----------------
// Block = 256 threads = 8 waves; wave w computes the 16x16 tile at (blockIdx.x*16, w*16).
// A-layout (16x4 f32, ISA 7.12.2): lanes 0-15 hold M=lane, K=kk..kk+1; lanes 16-31 K=kk+2..kk+3.
// B-layout (4x16): lanes 0-15 hold N=lane rows K=kk..kk+1 (v0,v1); lanes 16-31 rows K=kk+2..kk+3.
// C/D (16x16 f32): VGPR r -> M = r + 8*(lane>=16), N = lane&15.
__global__ void gemm_bias_wmma(const float* __restrict__ A, const float* __restrict__ B,
                               const float* __restrict__ bias, float* __restrict__ C,
                               int M, int K) {
  const int lane = threadIdx.x & 31;
  const int wave = threadIdx.x >> 5;
  const int m0 = blockIdx.x << 4;
  const int n0 = wave << 4;
  const int half = lane >> 4;
  const int l = lane & 15;
  v8f acc = {};
  for (int kk = 0; kk < K; kk += 4) {
    const int ka = kk + (half << 1);
    v2f a, b;
    a[0] = A[(size_t)(m0 + l) * K + ka];
    a[1] = A[(size_t)(m0 + l) * K + ka + 1];
    b[0] = B[(size_t)ka * 128 + n0 + l];
    b[1] = B[(size_t)(ka + 1) * 128 + n0 + l];
    acc = __builtin_amdgcn_wmma_f32_16x16x4_f32(false, a, false, b, (short)0, acc,
                                                false, false);
  }
  const float bb = bias[n0 + l];
  const int rowBase = m0 + (half << 3);
#pragma unroll
  for (int r = 0; r < 8; ++r)
    C[(size_t)(rowBase + r) * 128 + n0 + l] = acc[r] + bb;
}

// ---------------- float atomic max (sign-aware int trick) ----------------
__device__ inline void atomicMaxF32(float* addr, float val) {
  if (val >= 0.0f)
    atomicMax((int*)addr, __float_as_int(val));
  else
    atomicMin((unsigned int*)addr, (unsigned int)__float_as_int(val));
}

// ---------------- edge pass 1: alpha + segment max ----------------
// One wave per edge; 4 channels per lane (128 total). Head h owns ch contiguous channels.
__global__ void edge_alpha_kernel(const int* __restrict__ src, const int* __restrict__ dst,
                                  const float* __restrict__ q, const float* __restrict__ k,
                                  const float* __restrict__ ek,
                                  float* __restrict__ alpha, float* __restrict__ mbuf,
                                  int E, int heads, int ch, float scale) {
  const int lane = threadIdx.x & 31;
  const int e = blockIdx.x * (blockDim.x >> 5) + (threadIdx.x >> 5);
  if (e >= E) return;
  const int s = src[e], d = dst[e];
  const float4 qv = ((const float4*)(q + (size_t)d * 128))[lane];
  const float4 kv = ((const float4*)(k + (size_t)s * 128))[lane];
  const float4 ev = ((const float4*)(ek + (size_t)e * 128))[lane];
  float part = qv.x * (kv.x + ev.x) + qv.y * (kv.y + ev.y) +
               qv.z * (kv.z + ev.z) + qv.w * (kv.w + ev.w);
  const int lph = ch >> 2;  // lanes per head: 4 (ch=16) or 32 (ch=128)
  for (int off = 1; off < lph; off <<= 1)
    part += __shfl_xor(part, off, 32);
  if ((lane & (lph - 1)) == 0) {
    const int h = lane / lph;
    const float a = part * scale;
    alpha[(size_t)e * heads + h] = a;
    atomicMaxF32(&mbuf[(size_t)d * heads + h], a);
  }
}

// ---------------- edge pass 2: exp + fused weighted scatter + denom ----------------
__global__ void edge_scatter_kernel(const int* __restrict__ src, const int* __restrict__ dst,
                                    const float* __restrict__ v, const float* __restrict__ ek,
                                    const float* __restrict__ alpha, const float* __restrict__ mbuf,
                                    float* __restrict__ num, float* __restrict__ den,
                                    int E, int heads, int ch) {
  const int lane = threadIdx.x & 31;
  const int e = blockIdx.x * (blockDim.x >> 5) + (threadIdx.x >> 5);
  if (e >= E) return;
  const int s = src[e], d = dst[e];
  const int lph = ch >> 2;
  const int h = lane / lph;
  const float a = alpha[(size_t)e * heads + h];
  const float m = mbuf[(size_t)d * heads + h];
  const float ex = __expf(a - m);
  const float4 vv = ((const float4*)(v + (size_t)s * 128))[lane];
  const float4 ev = ((const float4*)(ek + (size_t)e * 128))[lane];
  float* np = num + (size_t)d * 128 + lane * 4;
  atomicAdd(np + 0, (vv.x + ev.x) * ex);
  atomicAdd(np + 1, (vv.y + ev.y) * ex);
  atomicAdd(np + 2, (vv.z + ev.z) * ex);
  atomicAdd(np + 3, (vv.w + ev.w) * ex);
  if ((lane & (lph - 1)) == 0)
    atomicAdd(&den[(size_t)d * heads + h], ex);
}

// ---------------- node finalize: out = relu(num/(den+eps) + skip) ----------------
__global__ void finalize_kernel(const float* __restrict__ num, const float* __restrict__ den,
                                const float* __restrict__ skip, float* __restrict__ out,
                                int N, int heads, int ch) {
  const long i = (long)blockIdx.x * blockDim.x + threadIdx.x;
  if (i >= (long)N * 128) return;
  const int n = (int)(i >> 7);
  const int c = (int)(i & 127);
  const int h = c / ch;
  const float val = num[i] / (den[(size_t)n * heads + h] + 1e-16f) + skip[i];
  out[i] = fmaxf(val, 0.0f);
}

extern "C" void kernel_launch(void* const* d_in, const int* in_sizes, int n_in,
                              void* d_out, int out_size, void* d_ws, size_t ws_size,
                              hipStream_t stream) {
  const int N = NN, E = NE;
  const float* x = (const float*)d_in[0];
  const int* src = (const int*)d_in[1];       // edge_index[0]
  const int* dst = src + E;                    // edge_index[1]
  const float* ef = (const float*)d_in[2];

  // scratch layout (floats)
  float* W = (float*)d_ws;
  size_t off = 0;
  float* ek    = W + off; off += (size_t)E * 128;
  float* qb    = W + off; off += (size_t)N * 128;
  float* kb    = W + off; off += (size_t)N * 128;
  float* vb    = W + off; off += (size_t)N * 128;
  float* skipb = W + off; off += (size_t)N * 128;
  float* numb  = W + off; off += (size_t)N * 128;
  float* hbuf  = W + off; off += (size_t)N * 128;
  float* alpha = W + off; off += (size_t)E * 8;
  float* mbuf  = W + off; off += (size_t)N * 8;
  float* denb  = W + off; off += (size_t)N * 8;

  auto layer = [&](const float* xin, void* const* p, int heads, int ch, float* outp) {
    const float* Wq = (const float*)p[0]; const float* bq = (const float*)p[1];
    const float* Wk = (const float*)p[2]; const float* bk = (const float*)p[3];
    const float* Wv = (const float*)p[4]; const float* bv = (const float*)p[5];
    const float* We = (const float*)p[6]; const float* be = (const float*)p[7];
    const float* Ws = (const float*)p[8]; const float* bs = (const float*)p[9];

    gemm_bias_wmma<<<N / 16, 256, 0, stream>>>(xin, Wq, bq, qb,    N, 128);
    gemm_bias_wmma<<<N / 16, 256, 0, stream>>>(xin, Wk, bk, kb,    N, 128);
    gemm_bias_wmma<<<N / 16, 256, 0, stream>>>(xin, Wv, bv, vb,    N, 128);
    gemm_bias_wmma<<<N / 16, 256, 0, stream>>>(xin, Ws, bs, skipb, N, 128);
    gemm_bias_wmma<<<E / 16, 256, 0, stream>>>(ef,  We, be, ek,    E, 64);

    const long nmh = (long)N * heads;
    const long nnum = (long)N * 128;
    fill_kernel<<<(int)((nmh  + 255) / 256), 256, 0, stream>>>(mbuf, -INFINITY, nmh);
    fill_kernel<<<(int)((nmh  + 255) / 256), 256, 0, stream>>>(denb, 0.0f, nmh);
    fill_kernel<<<(int)((nnum + 255) / 256), 256, 0, stream>>>(numb, 0.0f, nnum);

    const float scale = 1.0f / sqrtf((float)ch);
    const int epb = 8;  // 8 waves (edges) per 256-thread block
    edge_alpha_kernel<<<(E + epb - 1) / epb, 256, 0, stream>>>(
        src, dst, qb, kb, ek, alpha, mbuf, E, heads, ch, scale);
    edge_scatter_kernel<<<(E + epb - 1) / epb, 256, 0, stream>>>(
        src, dst, vb, ek, alpha, mbuf, numb, denb, E, heads, ch);
    finalize_kernel<<<(int)((nnum + 255) / 256), 256, 0, stream>>>(
        numb, denb, skipb, outp, N, heads, ch);
  };

  // layer 1: heads=8, ch=16, output -> hbuf; layer 2: heads=1, ch=128 -> d_out
  layer(x,    d_in + 3,  8, 16,  hbuf);
  layer(hbuf, d_in + 13, 1, 128, (float*)d_out);
}